// VQ_80702435492321
// MI455X (gfx1250) — compile-verified
//
#include <hip/hip_runtime.h>
#include <hip/hip_bf16.h>

typedef float v2f __attribute__((ext_vector_type(2)));
typedef float v8f __attribute__((ext_vector_type(8)));

#define S_CODES   1024
#define D_DIM     128
#define T_DIM     2048
#define B_DIM     16
#define N_ROWS    (B_DIM * T_DIM)      // 32768
#define EMA_F     0.99f
#define ONE_M_EMA 0.01f

// ---------------------------------------------------------------------------
// Prep: pack embedding into WMMA-B pair layout embT2[k/2][s][2] (so one lane's
// (K,K+1) pair is a single b64), compute e2[s]=||E_s||^2, zero accumulators.
// ---------------------------------------------------------------------------
__global__ __launch_bounds__(128) void vq_prep(const float* __restrict__ emb,
                                               float* __restrict__ embT2,
                                               float* __restrict__ e2,
                                               float* __restrict__ counts,
                                               float* __restrict__ diff_acc,
                                               float* __restrict__ sums) {
  const int s = blockIdx.x;     // 0..1023
  const int d = threadIdx.x;    // 0..127 (== k)
  const float v = emb[s * D_DIM + d];
  embT2[(d >> 1) * (S_CODES * 2) + s * 2 + (d & 1)] = v;
  sums[s * D_DIM + d] = 0.0f;

  __shared__ float sh[128];
  sh[d] = v * v;
  __syncthreads();
  for (int st = 64; st > 0; st >>= 1) {
    if (d < st) sh[d] += sh[d + st];
    __syncthreads();
  }
  if (d == 0) {
    e2[s] = sh[0];
    counts[s] = 0.0f;
    if (s == 0) *diff_acc = 0.0f;
  }
}

// ---------------------------------------------------------------------------
// Main: block = 16 consecutive tokens (one b). 8 waves; each wave scans 128
// codes with V_WMMA_F32_16X16X4_F32. B tiles are double-buffered in registers
// so the 32-WMMA chain per tile runs behind a single load wait.
// ---------------------------------------------------------------------------
__global__ __launch_bounds__(256) void vq_main(const float* __restrict__ x,
                                               const float* __restrict__ emb,
                                               const float* __restrict__ embT2,
                                               const float* __restrict__ e2,
                                               float* __restrict__ out_q,
                                               float* __restrict__ sums,
                                               float* __restrict__ counts,
                                               float* __restrict__ diff_acc) {
  __shared__ float lds_x[16 * D_DIM];        // A tile, [kpair][m][2] layout
  __shared__ float red_v[8 * 16 * 16];       // argmin values (reused for diff reduce)
  __shared__ int   red_i[8 * 16 * 16];       // argmin indices
  __shared__ int   codes[16];

  const int tid   = threadIdx.x;
  const int blk   = blockIdx.x;              // 0..2047
  const int b     = blk >> 7;
  const int t0    = (blk & 127) << 4;
  const float* xb = x + (size_t)b * D_DIM * T_DIM;

  for (int i = tid; i < 16 * D_DIM; i += 256) {
    const int d = i >> 4, m = i & 15;
    lds_x[(d >> 1) * 32 + m * 2 + (d & 1)] = xb[(size_t)d * T_DIM + t0 + m];
  }
  __syncthreads();

  const int wave = tid >> 5, lane = tid & 31;
  const int half = lane >> 4, lc = lane & 15;

  // A tile resident in 64 VGPRs: lane (half,lc) holds rows M=lc, K pair
  // {4ks+2*half, +1} for each of 32 K-steps.
  v2f a[32];
#pragma unroll
  for (int ks = 0; ks < 32; ++ks) {
    const int drow = 4 * ks + 2 * half;
    a[ks] = *(const v2f*)&lds_x[(drow >> 1) * 32 + lc * 2];
  }

  // B base for this lane: element (nt, ks) at bp[ks*2048 + nt*16] (v2f units).
  const v2f* bp = (const v2f*)(embT2 + half * (S_CODES * 2) + (wave * 128 + lc) * 2);

  float minv[8];
  int   mini[8];
#pragma unroll
  for (int r = 0; r < 8; ++r) { minv[r] = 3.4e38f; mini[r] = 0; }

  v2f breg[2][32];
#pragma unroll
  for (int ks = 0; ks < 32; ++ks) breg[0][ks] = bp[ks * 2048];

#pragma unroll
  for (int nt = 0; nt < 8; ++nt) {
    const int cur = nt & 1;
    if (nt < 7) {
#pragma unroll
      for (int ks = 0; ks < 32; ++ks)
        breg[cur ^ 1][ks] = bp[ks * 2048 + (nt + 1) * 16];
    }
    v8f acc = {};
#pragma unroll
    for (int ks = 0; ks < 32; ++ks)
      acc = __builtin_amdgcn_wmma_f32_16x16x4_f32(
          false, a[ks], false, breg[cur][ks], (short)0, acc, false, false);

    const int n0 = wave * 128 + nt * 16;
    const float e2v = e2[n0 + lc];
#pragma unroll
    for (int r = 0; r < 8; ++r) {
      const float v = e2v - 2.0f * acc[r];     // C: VGPR r = row (r+8*half), col lc
      if (v < minv[r]) { minv[r] = v; mini[r] = n0 + lc; }
    }
  }

#pragma unroll
  for (int r = 0; r < 8; ++r) {
    const int m = r + 8 * half;
    red_v[wave * 256 + m * 16 + lc] = minv[r];
    red_i[wave * 256 + m * 16 + lc] = mini[r];
  }
  __syncthreads();

  if (tid < 16) {
    const int m = tid;
    float best = 3.4e38f;
    int   bi   = 0x7fffffff;
    for (int w = 0; w < 8; ++w)
      for (int j = 0; j < 16; ++j) {
        const float v = red_v[w * 256 + m * 16 + j];
        const int   i = red_i[w * 256 + m * 16 + j];
        if (v < best || (v == best && i < bi)) { best = v; bi = i; }
      }
    codes[m] = bi;
  }
  __syncthreads();

  // Outputs: quantized (coalesced along t), diff partial, counts, sums.
  const int m = tid & 15;
  const int c = codes[m];
  float dacc = 0.0f;
  float* outq = out_q + (size_t)b * D_DIM * T_DIM + t0 + m;
#pragma unroll
  for (int jj = 0; jj < 8; ++jj) {
    const int d  = (tid >> 4) + 16 * jj;
    const float qv = emb[c * D_DIM + d];
    outq[(size_t)d * T_DIM] = qv;
    const float xv = lds_x[(d >> 1) * 32 + m * 2 + (d & 1)];
    const float dd = xv - qv;
    dacc += dd * dd;
    atomicAdd(&sums[c * D_DIM + d], xv);
  }

  red_v[tid] = dacc;
  __syncthreads();
  for (int st = 128; st > 0; st >>= 1) {
    if (tid < st) red_v[tid] += red_v[tid + st];
    __syncthreads();
  }
  if (tid == 0) atomicAdd(diff_acc, red_v[0]);
  if (tid < 16) atomicAdd(&counts[codes[tid]], 1.0f);
}

// ---------------------------------------------------------------------------
// Finalize: EMA blend. out_emb currently holds per-code sums (in place).
// ---------------------------------------------------------------------------
__global__ __launch_bounds__(128) void vq_fin(const float* __restrict__ emb,
                                              const float* __restrict__ usage,
                                              const float* __restrict__ counts,
                                              const float* __restrict__ diff_acc,
                                              float* __restrict__ out_emb,
                                              float* __restrict__ out_usage,
                                              float* __restrict__ out_diff) {
  const int s = blockIdx.x;
  const int d = threadIdx.x;
  const float cnt = counts[s];
  const float sum = out_emb[s * D_DIM + d];
  const float e   = emb[s * D_DIM + d];
  const float target = (cnt > 0.0f) ? (sum / fmaxf(cnt, 1.0f)) : e;
  out_emb[s * D_DIM + d] = EMA_F * e + ONE_M_EMA * target;
  if (d == 0) {
    out_usage[s] = EMA_F * usage[s] + ONE_M_EMA * cnt;
    if (s == 0) *out_diff = *diff_acc * (1.0f / (float)(N_ROWS * (size_t)D_DIM));
  }
}

extern "C" void kernel_launch(void* const* d_in, const int* in_sizes, int n_in,
                              void* d_out, int out_size, void* d_ws, size_t ws_size,
                              hipStream_t stream) {
  (void)in_sizes; (void)n_in; (void)out_size; (void)ws_size;
  const float* x     = (const float*)d_in[0];   // [16,128,2048]
  const float* emb   = (const float*)d_in[1];   // [1024,128]
  const float* usage = (const float*)d_in[2];   // [1024]

  float* out = (float*)d_out;
  const size_t Q = (size_t)B_DIM * D_DIM * T_DIM;      // 4,194,304
  float* out_diff  = out + Q;
  float* out_emb   = out + Q + 1;                      // 131,072 (doubles as sums acc)
  float* out_usage = out + Q + 1 + S_CODES * D_DIM;    // 1,024

  float* ws       = (float*)d_ws;
  float* embT2    = ws;                                // 131,072 (pair-packed)
  float* e2       = ws + 131072;                       // 1,024
  float* counts   = ws + 132096;                       // 1,024
  float* diff_acc = ws + 133120;                       // 1

  vq_prep<<<S_CODES, 128, 0, stream>>>(emb, embT2, e2, counts, diff_acc, out_emb);
  vq_main<<<N_ROWS / 16, 256, 0, stream>>>(x, emb, embT2, e2, out, out_emb, counts, diff_acc);
  vq_fin<<<S_CODES, 128, 0, stream>>>(emb, usage, counts, diff_acc, out_emb, out_usage, out_diff);
}